// AttZipLlamaAttention_37666863186508
// MI455X (gfx1250) — compile-verified
//
#include <hip/hip_runtime.h>
#include <hip/hip_bf16.h>

typedef __attribute__((ext_vector_type(2))) float v2f;
typedef __attribute__((ext_vector_type(8))) float v8f;

#define SEQ 4096
#define HEADS 32
#define HD 128
#define RECENT 512
#define IMP 4
#define KEEP (IMP + RECENT)          // 516
#define NROWS (HEADS * SEQ)          // 131072 rows to reduce over
#define NCHUNK 32                    // row chunks (partial buffers)
#define ROWS_PER_CHUNK (NROWS / NCHUNK) // 4096

// ---------------------------------------------------------------------------
// Kernel 1: column-sum partials via V_WMMA_F32_16X16X4_F32 with B = ones.
// A tile: M(=16 attn columns) x K(=4 attn rows). D[m][n] = sum_k A[m][k].
// A-layout (ISA 7.12.2, 32-bit A 16x4): lane<16 -> M=lane, VGPR0=K0, VGPR1=K1;
// lane>=16 -> M=lane-16, VGPR0=K2, VGPR1=K3.
// D-layout: lane<16: VGPRd = M=d ; lane>=16: VGPRd = M=d+8 (identical over N).
// ---------------------------------------------------------------------------
__global__ __launch_bounds__(256) void reduce_partial_kernel(
    const float* __restrict__ attn, float* __restrict__ partial) {
  const int lane = threadIdx.x & 31;
  const int wave = threadIdx.x >> 5;                 // 0..7
  const int j0   = blockIdx.x * 128 + wave * 16;     // column base for this wave
  const int m    = lane & 15;
  const int half = lane >> 4;                        // 0: K=0,1  1: K=2,3
  const long rowBase = (long)blockIdx.y * ROWS_PER_CHUNK;
  const float* base = attn + rowBase * SEQ + j0 + m;

  v2f bones; bones.x = 1.0f; bones.y = 1.0f;         // B = ones (4x16)
  v8f acc0 = {};
  v8f acc1 = {};

  for (int i = 0; i < ROWS_PER_CHUNK; i += 8) {
    long r0 = (long)(i + 2 * half) * SEQ;
    v2f a0, a1;
    a0.x = __builtin_nontemporal_load(base + r0);            // row i+2h
    a0.y = __builtin_nontemporal_load(base + r0 + SEQ);      // row i+2h+1
    a1.x = __builtin_nontemporal_load(base + r0 + 4 * SEQ);  // row i+4+2h
    a1.y = __builtin_nontemporal_load(base + r0 + 5 * SEQ);  // row i+4+2h+1
    acc0 = __builtin_amdgcn_wmma_f32_16x16x4_f32(false, a0, false, bones,
                                                 (short)0, acc0, false, false);
    acc1 = __builtin_amdgcn_wmma_f32_16x16x4_f32(false, a1, false, bones,
                                                 (short)0, acc1, false, false);
  }

  // lane 0 holds sums for columns j0+0..7 in acc[0..7]; lane 16 for j0+8..15.
  if (m == 0) {
    float* p = partial + (size_t)blockIdx.y * SEQ + j0 + half * 8;
#pragma unroll
    for (int d = 0; d < 8; ++d) p[d] = acc0[d] + acc1[d];
  }
}

// ---------------------------------------------------------------------------
// Kernel 2: fold NCHUNK partials, divide by num_heads -> imp_score[4096]
// ---------------------------------------------------------------------------
__global__ void reduce_final_kernel(const float* __restrict__ partial,
                                    float* __restrict__ imp) {
  int j = blockIdx.x * blockDim.x + threadIdx.x;     // 0..4095
  float s = 0.0f;
#pragma unroll
  for (int c = 0; c < NCHUNK; ++c) s += partial[(size_t)c * SEQ + j];
  imp[j] = s * (1.0f / (float)HEADS);
}

// ---------------------------------------------------------------------------
// Kernel 3: top-4 of imp[0:3584] (lowest-index tie-break, like lax.top_k),
// sort ascending, build keep_idx, emit imp_out / counter_out.
// ---------------------------------------------------------------------------
__global__ void topk_kernel(const float* __restrict__ imp,
                            int* __restrict__ keep_idx,
                            float* __restrict__ out_imp,
                            float* __restrict__ out_counter) {
  __shared__ float svals[256];
  __shared__ int   sidx[256];
  __shared__ int   chosen[IMP];
  const int tid = threadIdx.x;
  const int Nsel = SEQ - RECENT;                     // 3584

  for (int r = 0; r < IMP; ++r) {
    float best = -__builtin_huge_valf();
    int   bidx = 0x7fffffff;
    for (int j = tid; j < Nsel; j += 256) {
      bool skip = false;
      for (int p = 0; p < r; ++p) if (chosen[p] == j) skip = true;
      float v = imp[j];
      if (!skip && (v > best || (v == best && j < bidx))) { best = v; bidx = j; }
    }
    svals[tid] = best; sidx[tid] = bidx;
    __syncthreads();
    for (int s = 128; s > 0; s >>= 1) {
      if (tid < s) {
        float v2 = svals[tid + s]; int i2 = sidx[tid + s];
        if (v2 > svals[tid] || (v2 == svals[tid] && i2 < sidx[tid])) {
          svals[tid] = v2; sidx[tid] = i2;
        }
      }
      __syncthreads();
    }
    if (tid == 0) chosen[r] = sidx[0];
    __syncthreads();
  }

  if (tid == 0) {                                    // sort 4 ascending
    for (int a = 0; a < IMP; ++a)
      for (int b = a + 1; b < IMP; ++b)
        if (chosen[b] < chosen[a]) { int t = chosen[a]; chosen[a] = chosen[b]; chosen[b] = t; }
  }
  __syncthreads();

  for (int t = tid; t < KEEP; t += 256) {
    int j = (t < IMP) ? chosen[t] : (Nsel + (t - IMP));
    keep_idx[t]    = j;
    out_imp[t]     = imp[j];
    out_counter[t] = (float)(SEQ - j);
  }
}

// ---------------------------------------------------------------------------
// Kernel 4: gather K and V rows. One 32-lane wave per (token, head); each lane
// moves a float4 -> 512B contiguous per wave in and out.
// ---------------------------------------------------------------------------
__global__ void gather_kernel(const float* __restrict__ k_cache,
                              const float* __restrict__ v_cache,
                              const int* __restrict__ keep_idx,
                              float* __restrict__ k_out,
                              float* __restrict__ v_out) {
  const int t = blockIdx.x;        // 0..515
  const int h = blockIdx.y;        // 0..31
  const int lane = threadIdx.x;    // 0..31
  const int j = keep_idx[t];
  const float4* ksrc = (const float4*)(k_cache + ((size_t)h * SEQ + j) * HD);
  const float4* vsrc = (const float4*)(v_cache + ((size_t)h * SEQ + j) * HD);
  float4* kdst = (float4*)(k_out + ((size_t)h * KEEP + t) * HD);
  float4* vdst = (float4*)(v_out + ((size_t)h * KEEP + t) * HD);
  kdst[lane] = ksrc[lane];
  vdst[lane] = vsrc[lane];
}

// ---------------------------------------------------------------------------
extern "C" void kernel_launch(void* const* d_in, const int* in_sizes, int n_in,
                              void* d_out, int out_size, void* d_ws, size_t ws_size,
                              hipStream_t stream) {
  const float* k_cache = (const float*)d_in[0];   // (1,32,4096,128)
  const float* v_cache = (const float*)d_in[1];   // (1,32,4096,128)
  const float* attn    = (const float*)d_in[2];   // (1,32,4096,4096)

  float* out = (float*)d_out;
  const size_t KV_ELEMS = (size_t)HEADS * KEEP * HD;   // 2,113,536
  float* k_out       = out;
  float* v_out       = out + KV_ELEMS;
  float* imp_out     = out + 2 * KV_ELEMS;
  float* counter_out = out + 2 * KV_ELEMS + KEEP;

  // Workspace layout
  float* partial = (float*)d_ws;                                   // 32*4096 f32
  float* imp     = partial + (size_t)NCHUNK * SEQ;                 // 4096 f32
  int*   keepIdx = (int*)(imp + SEQ);                              // 516 i32

  // 1) 2.15 GB streaming column-sum via WMMA (bandwidth-bound, ~92us floor)
  reduce_partial_kernel<<<dim3(SEQ / 128, NCHUNK), 256, 0, stream>>>(attn, partial);
  // 2) fold partials + mean over heads
  reduce_final_kernel<<<SEQ / 256, 256, 0, stream>>>(partial, imp);
  // 3) top-4 + index build + small outputs
  topk_kernel<<<1, 256, 0, stream>>>(imp, keepIdx, imp_out, counter_out);
  // 4) K/V gather (~34 MB total traffic)
  gather_kernel<<<dim3(KEEP, HEADS), 32, 0, stream>>>(k_cache, v_cache, keepIdx,
                                                      k_out, v_out);
}